// TAGDN_74663711474280
// MI455X (gfx1250) — compile-verified
//
#include <hip/hip_runtime.h>

#define NNODES 100000
#define NEDGES 1600000
#define WIN    256
#define WHID   64
#define NTYPE  4
#define NH     (NNODES * WHID)   // 6,400,000
#define NPAD   (1 << 23)         // 8,388,608
#define KHOPS  10
#define ALPHA  0.1f

typedef __attribute__((ext_vector_type(2))) float v2f;
typedef __attribute__((ext_vector_type(8))) float v8f;

// ---------------- WMMA fp32 GEMM: H = X @ W + b ----------------
// block = 128 threads (4 waves), each block -> 16 rows x 64 cols.
// Wave w computes column tile [16w, 16w+16) with V_WMMA_F32_16X16X4_F32.
// 16x16x4 f32 fragment layout (wave32):
//   A (16x4): lane = {half,m}; VGPR0/1 = K={2*half, 2*half+1}, row M=m
//   B (4x16): lane = {half,m}; VGPR0/1 = K={2*half, 2*half+1}, col N=m
//   C (16x16): VGPR r: lanes 0-15 -> M=r, lanes 16-31 -> M=r+8; N=lane&15
__global__ __launch_bounds__(128) void encode_gemm(const float* __restrict__ X,
    const float* __restrict__ W, const float* __restrict__ bias,
    float* __restrict__ H)
{
  __shared__ __align__(16) float sA[16 * 128];    // 8 KB
  __shared__ __align__(16) float sB[128 * WHID];  // 32 KB
  const int tid      = threadIdx.x;
  const int row_base = blockIdx.x * 16;
  const int wave     = tid >> 5;
  const int lane     = tid & 31;
  const int half     = lane >> 4;
  const int m        = lane & 15;
  const int col_base = wave * 16;

  v8f c = {};
  for (int k0 = 0; k0 < WIN; k0 += 128) {
    for (int i = tid; i < 16 * 128 / 4; i += 128) {          // stage A 16x128
      int r = i >> 5, c4 = i & 31;
      ((float4*)sA)[i] = *(const float4*)(X + (size_t)(row_base + r) * WIN + k0 + c4 * 4);
    }
    for (int i = tid; i < 128 * WHID / 4; i += 128) {        // stage B 128x64
      int r = i >> 4, c4 = i & 15;
      ((float4*)sB)[i] = *(const float4*)(W + (size_t)(k0 + r) * WHID + c4 * 4);
    }
    __syncthreads();
    #pragma unroll
    for (int kk = 0; kk < 128; kk += 4) {
      v2f a, b;
      a.x = sA[m * 128 + kk + 2 * half];
      a.y = sA[m * 128 + kk + 2 * half + 1];
      b.x = sB[(kk + 2 * half) * WHID + col_base + m];
      b.y = sB[(kk + 2 * half + 1) * WHID + col_base + m];
      c = __builtin_amdgcn_wmma_f32_16x16x4_f32(false, a, false, b, (short)0, c, false, false);
    }
    __syncthreads();
  }
  const float bv = bias[col_base + m];
  #pragma unroll
  for (int r = 0; r < 8; ++r) {
    int row = row_base + r + 8 * half;
    H[(size_t)row * WHID + col_base + m] = c[r] + bv;
  }
}

// ---------------- helpers ----------------
__device__ __forceinline__ float waveSum(float v) {
  #pragma unroll
  for (int off = 16; off > 0; off >>= 1) v += __shfl_xor(v, off, 32);
  return v;
}

// ---------------- L2 row-normalize + per-type sums ----------------
__global__ __launch_bounds__(256) void norm_stats(float* __restrict__ H,
    const int* __restrict__ ntype, float* __restrict__ gsum, float* __restrict__ gcnt)
{
  __shared__ float lsum[NTYPE * WHID];
  __shared__ float lcnt[NTYPE];
  const int tid = threadIdx.x;
  for (int i = tid; i < NTYPE * WHID; i += 256) lsum[i] = 0.f;
  if (tid < NTYPE) lcnt[tid] = 0.f;
  __syncthreads();
  const int wave = tid >> 5, lane = tid & 31;
  const int row  = blockIdx.x * 8 + wave;   // N divisible by 8
  float2 v = *(float2*)(H + (size_t)row * WHID + lane * 2);
  float ss = waveSum(v.x * v.x + v.y * v.y);
  float inv = 1.0f / fmaxf(sqrtf(ss), 1e-12f);
  v.x *= inv; v.y *= inv;
  *(float2*)(H + (size_t)row * WHID + lane * 2) = v;
  const int t = ntype[row];
  atomicAdd(&lsum[t * WHID + lane * 2], v.x);
  atomicAdd(&lsum[t * WHID + lane * 2 + 1], v.y);
  if (lane == 0) atomicAdd(&lcnt[t], 1.0f);
  __syncthreads();
  for (int i = tid; i < NTYPE * WHID; i += 256) atomicAdd(&gsum[i], lsum[i]);
  if (tid < NTYPE) atomicAdd(&gcnt[tid], lcnt[tid]);
}

__global__ void stats_mean(const float* __restrict__ gsum,
                           const float* __restrict__ gcnt, float* __restrict__ mean) {
  int i = threadIdx.x;  // 256 threads
  mean[i] = gsum[i] / gcnt[i / WHID];
}

__global__ __launch_bounds__(256) void dev2_stats(const float* __restrict__ H,
    const int* __restrict__ ntype, const float* __restrict__ mean, float* __restrict__ gdev2)
{
  __shared__ float lsum[NTYPE * WHID];
  const int tid = threadIdx.x;
  for (int i = tid; i < NTYPE * WHID; i += 256) lsum[i] = 0.f;
  __syncthreads();
  const int wave = tid >> 5, lane = tid & 31;
  const int row  = blockIdx.x * 8 + wave;
  float2 v = *(const float2*)(H + (size_t)row * WHID + lane * 2);
  const int t = ntype[row];
  float2 mu = *(const float2*)(mean + t * WHID + lane * 2);
  float dx = v.x - mu.x, dy = v.y - mu.y;
  atomicAdd(&lsum[t * WHID + lane * 2], dx * dx);
  atomicAdd(&lsum[t * WHID + lane * 2 + 1], dy * dy);
  __syncthreads();
  for (int i = tid; i < NTYPE * WHID; i += 256) atomicAdd(&gdev2[i], lsum[i]);
}

__global__ void stats_std(const float* __restrict__ gdev2,
                          const float* __restrict__ gcnt, float* __restrict__ stdv) {
  int i = threadIdx.x;  // 256 threads
  stdv[i] = gdev2[i] / sqrtf(gcnt[i / WHID] - 1.0f);  // faithful to reference (sum/sqrt(n-1))
}

__global__ void make_tilde(const float* __restrict__ H, const int* __restrict__ ntype,
    const float* __restrict__ mean, const float* __restrict__ stdv, float* __restrict__ tl)
{
  int i = blockIdx.x * blockDim.x + threadIdx.x;  // NH threads
  int row = i >> 6, c = i & 63;
  int t = ntype[row];
  tl[i] = (H[i] - mean[t * WHID + c]) / stdv[t * WHID + c];
}

// ---------------- Wasserstein: pad, bitonic sort (L2-resident), |diff| mean ----------------
__global__ void pad_copy(const float* __restrict__ src, float* __restrict__ dst, int n) {
  int i = blockIdx.x * blockDim.x + threadIdx.x;  // NPAD threads
  dst[i] = (i < n) ? src[i] : 3.402823466e+38f;
}

__global__ void bitonic_step(float* __restrict__ S1, float* __restrict__ S2, int j, int k) {
  unsigned i = blockIdx.x * blockDim.x + threadIdx.x;
  float* S = blockIdx.y ? S2 : S1;
  unsigned ixj = i ^ (unsigned)j;
  if (ixj > i) {
    float a = S[i], b = S[ixj];
    bool up = ((i & (unsigned)k) == 0);
    if (up ? (a > b) : (a < b)) { S[i] = b; S[ixj] = a; }
  }
}

__global__ __launch_bounds__(256) void wdist(const float* __restrict__ A,
    const float* __restrict__ B, double* __restrict__ acc)
{
  __shared__ float red[256];
  int i = blockIdx.x * blockDim.x + threadIdx.x;  // NH threads
  red[threadIdx.x] = fabsf(A[i] - B[i]);
  __syncthreads();
  for (int s = 128; s > 0; s >>= 1) {
    if (threadIdx.x < s) red[threadIdx.x] += red[threadIdx.x + s];
    __syncthreads();
  }
  if (threadIdx.x == 0) atomicAdd(acc, (double)red[0]);
}

__global__ void loss_final(const double* __restrict__ acc, float* __restrict__ out) {
  out[NH] = (float)(*acc / (double)NH);
}

// ---------------- CSR-by-destination build (order within bucket irrelevant) ----------------
__global__ void deg_count(const int* __restrict__ dst, int* __restrict__ degi) {
  int e = blockIdx.x * blockDim.x + threadIdx.x;  // NEDGES threads
  atomicAdd(&degi[dst[e]], 1);
}

__global__ __launch_bounds__(1024) void scan_deg(const int* __restrict__ degi,
                                                 int* __restrict__ rowstart) {
  __shared__ int buf[1024];
  __shared__ int carry;
  int tid = threadIdx.x;
  if (tid == 0) carry = 0;
  __syncthreads();
  for (int base = 0; base < NNODES; base += 1024) {
    int x = (base + tid < NNODES) ? degi[base + tid] : 0;
    buf[tid] = x;
    __syncthreads();
    for (int off = 1; off < 1024; off <<= 1) {
      int t = (tid >= off) ? buf[tid - off] : 0;
      __syncthreads();
      buf[tid] += t;
      __syncthreads();
    }
    if (base + tid < NNODES) rowstart[base + tid] = carry + buf[tid] - x;  // exclusive
    __syncthreads();
    if (tid == 1023) carry += buf[1023];
    __syncthreads();
  }
  if (tid == 0) rowstart[NNODES] = NEDGES;
}

__global__ void fill_csr(const int* __restrict__ src, const int* __restrict__ dst,
    const int* __restrict__ rowstart, int* __restrict__ cursor, int* __restrict__ colidx)
{
  int e = blockIdx.x * blockDim.x + threadIdx.x;  // NEDGES threads
  int d = dst[e];
  int p = atomicAdd(&cursor[d], 1);
  colidx[rowstart[d] + p] = src[e];
}

// ---------------- pull-style PPR hop: Z' = 0.9 * (1/deg) * sum Z[src] + 0.1 * tilde ----------------
__global__ __launch_bounds__(256) void ppr_hop(const float* __restrict__ Zin,
    const float* __restrict__ tl, const int* __restrict__ rowstart,
    const int* __restrict__ colidx, float* __restrict__ Zout)
{
  const int wave = threadIdx.x >> 5, lane = threadIdx.x & 31;
  const int n = blockIdx.x * 8 + wave;
  const int s0 = rowstart[n], s1 = rowstart[n + 1];
  float accx = 0.f, accy = 0.f;
  for (int e = s0; e < s1; ++e) {
    int s = colidx[e];
    float2 z = *(const float2*)(Zin + (size_t)s * WHID + lane * 2);
    accx += z.x; accy += z.y;
  }
  float invd = 1.0f / (float)((s1 - s0) > 1 ? (s1 - s0) : 1);
  float2 t = *(const float2*)(tl + (size_t)n * WHID + lane * 2);
  float2 o;
  o.x = (1.0f - ALPHA) * accx * invd + ALPHA * t.x;
  o.y = (1.0f - ALPHA) * accy * invd + ALPHA * t.y;
  *(float2*)(Zout + (size_t)n * WHID + lane * 2) = o;
}

// ---------------- de-normalize + L2 row-normalize -> d_out ----------------
__global__ __launch_bounds__(256) void denorm_final(const float* __restrict__ Z,
    const int* __restrict__ ntype, const float* __restrict__ mean,
    const float* __restrict__ stdv, float* __restrict__ out)
{
  const int wave = threadIdx.x >> 5, lane = threadIdx.x & 31;
  const int row = blockIdx.x * 8 + wave;
  const int t = ntype[row];
  float2 z  = *(const float2*)(Z + (size_t)row * WHID + lane * 2);
  float2 sd = *(const float2*)(stdv + t * WHID + lane * 2);
  float2 mu = *(const float2*)(mean + t * WHID + lane * 2);
  float vx = z.x * sd.x + mu.x;
  float vy = z.y * sd.y + mu.y;
  float ss = waveSum(vx * vx + vy * vy);
  float inv = 1.0f / fmaxf(sqrtf(ss), 1e-12f);
  float2 o; o.x = vx * inv; o.y = vy * inv;
  *(float2*)(out + (size_t)row * WHID + lane * 2) = o;
}

extern "C" void kernel_launch(void* const* d_in, const int* in_sizes, int n_in,
                              void* d_out, int out_size, void* d_ws, size_t ws_size,
                              hipStream_t stream)
{
  (void)in_sizes; (void)n_in; (void)out_size; (void)ws_size;
  const float* X     = (const float*)d_in[0];
  const float* W     = (const float*)d_in[1];
  const float* b     = (const float*)d_in[2];
  const float* gauss = (const float*)d_in[3];
  const int*   ei    = (const int*)d_in[4];
  const int*   ntype = (const int*)d_in[5];
  const int* esrc = ei;
  const int* edst = ei + NEDGES;

  float* ws    = (float*)d_ws;
  float* H     = ws;
  float* tl    = ws + (size_t)NH;
  float* Za    = ws + (size_t)2 * NH;
  float* Zb    = ws + (size_t)3 * NH;
  float* S1    = ws + (size_t)4 * NH;
  float* S2    = S1 + (size_t)NPAD;
  float* stats = S2 + (size_t)NPAD;      // element offset 42,377,216 (even -> 8B aligned)
  float*  gsum  = stats;                 // 256
  float*  gdev2 = stats + 256;           // 256
  float*  mean  = stats + 512;           // 256
  float*  stdv  = stats + 768;           // 256
  float*  gcnt  = stats + 1024;          // 4
  double* lacc  = (double*)(stats + 1030);
  int* iws      = (int*)(stats + 1032);
  int* degi     = iws;
  int* cursor   = iws + NNODES;
  int* rowstart = iws + 2 * NNODES;      // N+1
  int* colidx   = iws + 3 * NNODES + 1;  // E

  hipMemsetAsync(stats, 0, 1032 * sizeof(float), stream);     // gsum/gdev2/mean/stdv/cnt/lacc
  hipMemsetAsync(degi, 0, 2 * (size_t)NNODES * sizeof(int), stream);  // degi + cursor

  // encode + stats + tilde
  encode_gemm<<<NNODES / 16, 128, 0, stream>>>(X, W, b, H);
  norm_stats<<<NNODES / 8, 256, 0, stream>>>(H, ntype, gsum, gcnt);
  stats_mean<<<1, 256, 0, stream>>>(gsum, gcnt, mean);
  dev2_stats<<<NNODES / 8, 256, 0, stream>>>(H, ntype, mean, gdev2);
  stats_std<<<1, 256, 0, stream>>>(gdev2, gcnt, stdv);
  make_tilde<<<NH / 256, 256, 0, stream>>>(H, ntype, mean, stdv, tl);

  // Wasserstein loss (both sort arrays fit in MI455X 192MB L2)
  pad_copy<<<NPAD / 256, 256, 0, stream>>>(gauss, S1, NH);
  pad_copy<<<NPAD / 256, 256, 0, stream>>>(tl, S2, NH);
  dim3 bg(NPAD / 256, 2);
  for (int k = 2; k <= NPAD; k <<= 1)
    for (int j = k >> 1; j > 0; j >>= 1)
      bitonic_step<<<bg, 256, 0, stream>>>(S1, S2, j, k);
  wdist<<<NH / 256, 256, 0, stream>>>(S1, S2, lacc);
  loss_final<<<1, 1, 0, stream>>>(lacc, (float*)d_out);

  // CSR by destination
  deg_count<<<NEDGES / 256, 256, 0, stream>>>(edst, degi);
  scan_deg<<<1, 1024, 0, stream>>>(degi, rowstart);
  fill_csr<<<NEDGES / 256, 256, 0, stream>>>(esrc, edst, rowstart, cursor, colidx);

  // K=10 pull-style diffusion hops (no atomics)
  const float* zin = tl;
  float* zout = Za;
  for (int h = 0; h < KHOPS; ++h) {
    ppr_hop<<<NNODES / 8, 256, 0, stream>>>(zin, tl, rowstart, colidx, zout);
    zin = zout;
    zout = (zout == Za) ? Zb : Za;
  }
  denorm_final<<<NNODES / 8, 256, 0, stream>>>(zin, ntype, mean, stdv, (float*)d_out);
}